// Word2ManSkipGramNegSampling_60833916781032
// MI455X (gfx1250) — compile-verified
//
#include <hip/hip_runtime.h>

typedef float v2f __attribute__((ext_vector_type(2)));
typedef float v8f __attribute__((ext_vector_type(8)));

#define EMBED 128

// LDS holds metric pre-swizzled into WMMA f32 16x16x4 B-fragment order:
//   sB dword index (k*512 + n*4 + hi*2 + par) = metric[4k + 2*hi + par][n]
// A lane (n = row, half = hi) reads its B fragment for K-chunk k as one
// contiguous 8B load; all 32 lanes of a b64 access cover all 64 banks once.
__global__ __launch_bounds__(256)
void w2m_bilinear_kernel(const int* __restrict__ xs,
                         const int* __restrict__ ys,
                         const float* __restrict__ metric,
                         const float* __restrict__ EM,
                         const float* __restrict__ NM,
                         float* __restrict__ out,
                         int batch)
{
    __shared__ float sB[32 * 512];   // 64 KB, no padding needed

    const int tid = threadIdx.x;

    // ---- cooperative load + swizzle: metric -> B-fragment order in LDS ----
    {
        const float4* g = (const float4*)metric;
        for (int i = tid; i < 128 * 32; i += 256) {   // 32 float4 per row
            int r   = i >> 5;
            int c4  = i & 31;
            float4 v = g[i];
            int k   = r >> 2;
            int hi  = (r >> 1) & 1;
            int par = r & 1;
            float* dst = &sB[k * 512 + hi * 2 + par + (c4 << 4)];
            dst[0]  = v.x;   // column c4*4+0
            dst[4]  = v.y;   // column c4*4+1
            dst[8]  = v.z;
            dst[12] = v.w;
        }
    }
    __syncthreads();

    const int lane = tid & 31;
    const int row  = lane & 15;     // A: M index / B,C: N index
    const int hi   = lane >> 4;     // lane group
    const int koff = hi * 2;        // K sub-offset inside a 4-chunk (A layout)
    const int m0   = hi * 8;        // C rows held by this lane group

    const int wavesPerBlock = 256 / 32;
    const int globalWave = blockIdx.x * wavesPerBlock + (tid >> 5);
    const int totalWaves = gridDim.x * wavesPerBlock;
    const int nTiles = batch >> 4;

    for (int tile = globalWave; tile < nTiles; tile += totalWaves) {
        const int base = tile << 4;

        // ---- gather A fragments: this lane's half of 16 EMBEDM rows ----
        // A layout (16x4 f32): lane<16 -> K = 4k+{0,1}; lane>=16 -> K = 4k+{2,3}
        const int xrow = xs[base + row];
        const float* erow = EM + (size_t)xrow * EMBED;
        v2f afrag[32];
        #pragma unroll
        for (int k = 0; k < 32; ++k)
            afrag[k] = *(const v2f*)(erow + 4 * k + koff);   // 8B aligned

        // ys rows needed for this lane's C rows
        const float* nrow[8];
        #pragma unroll
        for (int j = 0; j < 8; ++j)
            nrow[j] = NM + (size_t)ys[base + m0 + j] * EMBED + row;

        float dacc[8];
        #pragma unroll
        for (int j = 0; j < 8; ++j) dacc[j] = 0.0f;

        // ---- m = e @ metric: 4 iterations x (2 N-tiles, 32 K-steps each) ----
        // Two independent WMMA accumulator chains; every B fragment is a single
        // 8B LDS load off one shared base with immediate offsets (k*2048 + {0,256}).
        #pragma unroll 1
        for (int nt = 0; nt < 8; nt += 2) {
            const int n0 = nt << 4;
            const float* bbase = &sB[(n0 + row) * 4 + hi * 2];
            v8f c0 = {};
            v8f c1 = {};
            #pragma unroll
            for (int k = 0; k < 32; ++k) {
                v2f b0 = *(const v2f*)(bbase + k * 512);        // N-tile nt
                v2f b1 = *(const v2f*)(bbase + k * 512 + 64);   // N-tile nt+1
                c0 = __builtin_amdgcn_wmma_f32_16x16x4_f32(
                         false, afrag[k], false, b0, (short)0, c0, false, false);
                c1 = __builtin_amdgcn_wmma_f32_16x16x4_f32(
                         false, afrag[k], false, b1, (short)0, c1, false, false);
            }
            // C layout: lane holds m[m0+j][n0+row] in c0[j], m[m0+j][n0+16+row]
            // in c1[j]; fuse the dot with N[ys] directly from the fragments.
            #pragma unroll
            for (int j = 0; j < 8; ++j) {
                float nv0 = nrow[j][n0];
                float nv1 = nrow[j][n0 + 16];
                dacc[j] = fmaf(c0[j], nv0, dacc[j]);
                dacc[j] = fmaf(c1[j], nv1, dacc[j]);
            }
        }

        // ---- reduce each row's partial dot over the 16 lanes of the group ----
        #pragma unroll
        for (int j = 0; j < 8; ++j) {
            float v = dacc[j];
            v += __shfl_xor(v, 1, 32);
            v += __shfl_xor(v, 2, 32);
            v += __shfl_xor(v, 4, 32);
            v += __shfl_xor(v, 8, 32);
            dacc[j] = v;
        }

        // ---- sigmoid(2d - 1); one writer lane per output row ----
        #pragma unroll
        for (int j = 0; j < 8; ++j) {
            if (row == j) {
                float d = dacc[j];
                out[base + m0 + j] = 1.0f / (1.0f + __expf(1.0f - 2.0f * d));
            }
        }
    }
}

extern "C" void kernel_launch(void* const* d_in, const int* in_sizes, int n_in,
                              void* d_out, int out_size, void* d_ws, size_t ws_size,
                              hipStream_t stream) {
    const int*   xs     = (const int*)d_in[0];
    const int*   ys     = (const int*)d_in[1];
    const float* metric = (const float*)d_in[2];
    const float* EM     = (const float*)d_in[3];
    const float* NM     = (const float*)d_in[4];
    float*       out    = (float*)d_out;

    const int batch  = in_sizes[0];          // 262144
    const int blocks = 512;                  // 4096 waves -> 4 tiles/wave, metric
                                             // LDS load+swizzle amortized over 32 tiles
    hipLaunchKernelGGL(w2m_bilinear_kernel, dim3(blocks), dim3(256), 0, stream,
                       xs, ys, metric, EM, NM, out, batch);
}